// GCN_45689862095124
// MI455X (gfx1250) — compile-verified
//
#include <hip/hip_runtime.h>

// ---------------------------------------------------------------------------
// GCN forward for MI455X (gfx1250, wave32, WMMA).
//  - GEMMs: V_WMMA_F32_16X16X4_F32, B staged in LDS via async global->LDS
//    copy (GLOBAL_LOAD_ASYNC_TO_LDS_B128, waited with s_wait_asynccnt).
//  - Edge aggregation: GLOBAL_ATOMIC_ADD_F32 into L2-resident AGG.
//  - Mean pool: two-stage LDS (ds_add_f32) -> global atomic reduction.
// ---------------------------------------------------------------------------

#define N_NODES   50000
#define N_EDGES   800000
#define IN_FEAT   128
#define HIDDEN    64
#define N_CLASSES 10
#define N_GRAPHS  64

typedef float v2f __attribute__((ext_vector_type(2)));
typedef float v8f __attribute__((ext_vector_type(8)));
typedef int   v4i __attribute__((ext_vector_type(4)));

#if defined(__gfx1250__) && \
    __has_builtin(__builtin_amdgcn_global_load_async_to_lds_b128) && \
    __has_builtin(__builtin_amdgcn_s_wait_asynccnt)
#define GCN_HAS_ASYNC_LDS 1
typedef __attribute__((address_space(1))) v4i* gcn_gptr_t;   // global, v4i
typedef __attribute__((address_space(3))) v4i* gcn_lptr_t;   // LDS, v4i
#else
#define GCN_HAS_ASYNC_LDS 0
#endif

// ---- degree: deg[i] = 1 + in-degree(i); stored as dis = rsqrt(deg) ---------
__global__ void gcn_deg_count(const int* __restrict__ dst, float* __restrict__ deg) {
    int e = blockIdx.x * blockDim.x + threadIdx.x;
    if (e < N_EDGES) atomicAdd(&deg[dst[e]], 1.0f);
}

__global__ void gcn_deg_rsqrt(float* __restrict__ deg) {
    int i = blockIdx.x * blockDim.x + threadIdx.x;
    if (i < N_NODES) deg[i] = rsqrtf(1.0f + deg[i]);
}

// ---- fp32 WMMA GEMM: C[M x 64] = A[M x K] * B[K x 64], row-major -----------
// One wave per 16x16 output tile. blockDim = 128 (4 waves), wave w owns
// column tile w (N = 64 = 4 tiles); blockIdx.x owns row tile (M/16 = 3125).
// B is staged into LDS once per block (async copy when available).
template <int K>
__global__ void gcn_wmma_gemm(const float* __restrict__ A,
                              const float* __restrict__ B,
                              float* __restrict__ C) {
    __shared__ float Bs[K * HIDDEN];

    // --- stage B into LDS ---
    constexpr int totalVec = K * HIDDEN / 4;   // float4 chunks
#if GCN_HAS_ASYNC_LDS
    for (int i = threadIdx.x; i < totalVec; i += blockDim.x) {
        __builtin_amdgcn_global_load_async_to_lds_b128(
            (gcn_gptr_t)(B + (size_t)i * 4),
            (gcn_lptr_t)(Bs + (size_t)i * 4),
            /*offset=*/0, /*cpol=*/0);
    }
    __builtin_amdgcn_s_wait_asynccnt(0);
#else
    for (int i = threadIdx.x; i < totalVec; i += blockDim.x) {
        *reinterpret_cast<float4*>(Bs + (size_t)i * 4) =
            *reinterpret_cast<const float4*>(B + (size_t)i * 4);
    }
#endif
    __syncthreads();

    const int lane = threadIdx.x & 31;
    const int wave = threadIdx.x >> 5;
    const int half = lane >> 4;   // 0: lanes 0-15, 1: lanes 16-31
    const int l16  = lane & 15;
    const int row0 = blockIdx.x * 16;
    const int col0 = wave * 16;

    // A 16x4 layout: lanes 0-15 hold (M=l16, K=kb+0, kb+1); lanes 16-31 hold
    // (M=l16, K=kb+2, kb+3). B 4x16 mirrored across lanes.
    const float* __restrict__ arow = A + (size_t)(row0 + l16) * K;
    const float* bptr = Bs + col0 + l16;

    v8f acc = {};
#pragma unroll 4
    for (int kb = 0; kb < K; kb += 4) {
        const int k0 = kb + half * 2;
        v2f av, bv;
        av.x = arow[k0];
        av.y = arow[k0 + 1];
        bv.x = bptr[k0 * HIDDEN];
        bv.y = bptr[(k0 + 1) * HIDDEN];
        acc = __builtin_amdgcn_wmma_f32_16x16x4_f32(
            /*neg_a=*/false, av, /*neg_b=*/false, bv,
            /*c_mod=*/(short)0, acc, /*reuse_a=*/false, /*reuse_b=*/false);
    }

    // C/D layout: VGPR r -> row (r + 8*half), col l16.
    float* __restrict__ cp = C + (size_t)(row0 + half * 8) * HIDDEN + col0 + l16;
#pragma unroll
    for (int r = 0; r < 8; ++r) cp[(size_t)r * HIDDEN] = acc[r];
}

// ---- edge scatter: agg[dst] += h[src] * dis[src]*dis[dst] ------------------
// 16 threads per edge, float4 per thread (64 features).
__global__ void gcn_scatter(const float* __restrict__ H,
                            const int* __restrict__ src,
                            const int* __restrict__ dst,
                            const float* __restrict__ dis,
                            float* __restrict__ agg) {
    int t = blockIdx.x * blockDim.x + threadIdx.x;
    int e = t >> 4;
    if (e >= N_EDGES) return;
    int f4 = (t & 15) * 4;
    int s = src[e], d = dst[e];
    float norm = dis[s] * dis[d];
    const float4 hv = *reinterpret_cast<const float4*>(H + (size_t)s * HIDDEN + f4);
    float* ap = agg + (size_t)d * HIDDEN + f4;
    atomicAdd(ap + 0, hv.x * norm);
    atomicAdd(ap + 1, hv.y * norm);
    atomicAdd(ap + 2, hv.z * norm);
    atomicAdd(ap + 3, hv.w * norm);
}

// ---- self-loop + bias + relu ----------------------------------------------
__global__ void gcn_finalize(const float* __restrict__ H,
                             const float* __restrict__ agg,
                             const float* __restrict__ dis,
                             const float* __restrict__ bias,
                             float* __restrict__ act) {
    int t = blockIdx.x * blockDim.x + threadIdx.x;
    if (t >= N_NODES * HIDDEN) return;
    int i = t >> 6, f = t & 63;
    float di = dis[i];
    float v = agg[t] + H[t] * (di * di) + bias[f];
    act[t] = fmaxf(v, 0.0f);
}

// ---- global mean pool: LDS (ds_add_f32) partials -> global atomics ---------
#define POOL_NODES_PER_BLOCK 256
__global__ void gcn_pool(const float* __restrict__ act,
                         const int* __restrict__ batch,
                         float* __restrict__ sums,
                         float* __restrict__ counts) {
    __shared__ float lsum[N_GRAPHS * HIDDEN];   // 16 KB of the 320 KB WGP LDS
    __shared__ float lcnt[N_GRAPHS];

    for (int i = threadIdx.x; i < N_GRAPHS * HIDDEN; i += blockDim.x) lsum[i] = 0.0f;
    for (int i = threadIdx.x; i < N_GRAPHS; i += blockDim.x) lcnt[i] = 0.0f;
    __syncthreads();

    const int node0   = blockIdx.x * POOL_NODES_PER_BLOCK;
    const int nodeEnd = min(node0 + POOL_NODES_PER_BLOCK, N_NODES);
    const int f       = threadIdx.x & 63;          // feature lane
    for (int n = node0 + (threadIdx.x >> 6); n < nodeEnd; n += 4) {
        int g = batch[n];
        atomicAdd(&lsum[g * HIDDEN + f], act[(size_t)n * HIDDEN + f]);
        if (f == 0) atomicAdd(&lcnt[g], 1.0f);
    }
    __syncthreads();

    for (int i = threadIdx.x; i < N_GRAPHS * HIDDEN; i += blockDim.x) {
        float v = lsum[i];
        if (v != 0.0f) atomicAdd(&sums[i], v);
    }
    for (int i = threadIdx.x; i < N_GRAPHS; i += blockDim.x) {
        float v = lcnt[i];
        if (v != 0.0f) atomicAdd(&counts[i], v);
    }
}

// ---- dense head: out = (sums/counts) @ Wl + bl, [64 x 10] ------------------
__global__ void gcn_head(const float* __restrict__ sums,
                         const float* __restrict__ counts,
                         const float* __restrict__ Wl,
                         const float* __restrict__ bl,
                         float* __restrict__ out) {
    int t = blockIdx.x * blockDim.x + threadIdx.x;
    if (t >= N_GRAPHS * N_CLASSES) return;
    int g = t / N_CLASSES, c = t % N_CLASSES;
    float inv = 1.0f / fmaxf(counts[g], 1.0f);
    float acc = bl[c];
#pragma unroll 8
    for (int k = 0; k < HIDDEN; ++k)
        acc += sums[g * HIDDEN + k] * inv * Wl[k * N_CLASSES + c];
    out[t] = acc;
}

extern "C" void kernel_launch(void* const* d_in, const int* in_sizes, int n_in,
                              void* d_out, int out_size, void* d_ws, size_t ws_size,
                              hipStream_t stream) {
    (void)in_sizes; (void)n_in; (void)out_size; (void)ws_size;

    const float* x  = (const float*)d_in[0];                 // [50000,128]
    const int*   ei = (const int*)d_in[1];                   // [2,800000] (int32 per harness)
    const int*   bt = (const int*)d_in[2];                   // [50000]
    const float* W1 = (const float*)d_in[3];
    const float* b1 = (const float*)d_in[4];
    const float* W2 = (const float*)d_in[5];
    const float* b2 = (const float*)d_in[6];
    const float* W3 = (const float*)d_in[7];
    const float* b3 = (const float*)d_in[8];
    const float* Wl = (const float*)d_in[9];
    const float* bl = (const float*)d_in[10];
    float* out = (float*)d_out;

    const int* src = ei;
    const int* dst = ei + N_EDGES;

    // workspace layout (floats)
    float* ws     = (float*)d_ws;
    float* dis    = ws;                                  // 50000 (padded to 50048)
    float* H      = dis + 50048;                         // 50000*64
    float* AGG    = H   + (size_t)N_NODES * HIDDEN;      // 50000*64
    float* ACT    = AGG + (size_t)N_NODES * HIDDEN;      // 50000*64
    float* sums   = ACT + (size_t)N_NODES * HIDDEN;      // 64*64
    float* counts = sums + N_GRAPHS * HIDDEN;            // 64

    const int T256 = 256;
    const int nodeFeat = N_NODES * HIDDEN;

    // degrees -> dis = rsqrt(1 + in-degree)
    (void)hipMemsetAsync(dis, 0, N_NODES * sizeof(float), stream);
    gcn_deg_count<<<(N_EDGES + T256 - 1) / T256, T256, 0, stream>>>(dst, dis);
    gcn_deg_rsqrt<<<(N_NODES + T256 - 1) / T256, T256, 0, stream>>>(dis);

    auto layer = [&](const float* in, int K, const float* W, const float* b) {
        if (K == IN_FEAT)
            gcn_wmma_gemm<IN_FEAT><<<N_NODES / 16, 128, 0, stream>>>(in, W, H);
        else
            gcn_wmma_gemm<HIDDEN><<<N_NODES / 16, 128, 0, stream>>>(in, W, H);
        (void)hipMemsetAsync(AGG, 0, (size_t)nodeFeat * sizeof(float), stream);
        gcn_scatter<<<(N_EDGES * 16 + T256 - 1) / T256, T256, 0, stream>>>(H, src, dst, dis, AGG);
        gcn_finalize<<<(nodeFeat + T256 - 1) / T256, T256, 0, stream>>>(H, AGG, dis, b, ACT);
    };

    layer(x,   IN_FEAT, W1, b1);
    layer(ACT, HIDDEN,  W2, b2);
    layer(ACT, HIDDEN,  W3, b3);

    // mean pool + head
    (void)hipMemsetAsync(sums, 0, (N_GRAPHS * HIDDEN + N_GRAPHS) * sizeof(float), stream);
    gcn_pool<<<(N_NODES + POOL_NODES_PER_BLOCK - 1) / POOL_NODES_PER_BLOCK, T256, 0, stream>>>(
        ACT, bt, sums, counts);
    gcn_head<<<(N_GRAPHS * N_CLASSES + T256 - 1) / T256, T256, 0, stream>>>(sums, counts, Wl, bl, out);
}